// Model_13941463843473
// MI455X (gfx1250) — compile-verified
//
#include <hip/hip_runtime.h>
#include <math.h>

// ===========================================================================
// Bi-Mamba forecaster forward pass for MI455X (gfx1250, wave32).
// All GEMMs go through V_WMMA_F32_16X16X4_F32 (full f32 precision — the
// problem is ~25 GFLOP / ~100MB, L2-resident and latency-bound, so f32 WMMA
// is the right precision/throughput point; no benefit to bf16/fp8 here).
//
// GEMM contract (all call sites satisfy it):
//   * K % 4 == 0
//   * M % 64 == 0  (grid covers M exactly; A row index always in bounds)
//   * ragged N handled by clamping B's column (loads always in bounds,
//     stores guarded) -> branchless inner loop: b64 + 2x b32 loads + WMMA.
// ===========================================================================

typedef float v2f __attribute__((ext_vector_type(2)));
typedef float v8f __attribute__((ext_vector_type(8)));

#define SEQ_LEN   512
#define PRED_LEN  96
#define PATCH     24
#define STRIDE_   12
#define D_MODEL   128
#define D_STATE   16
#define D_INNER   256
#define DT_RANK   8
#define ENC_IN    32
#define BATCH     8
#define NUM_PATCH 41
#define NSERIES   256      /* BATCH*ENC_IN */
#define ROWS      10496    /* NSERIES*NUM_PATCH */
#define NFREQ     257
#define DBC_COLS  40       /* DT_RANK + 2*D_STATE */

#define ACT_NONE     0
#define ACT_SOFTPLUS 1
#define ACT_GELU     2
#define ACT_SIGMOID  3

__device__ __forceinline__ float apply_act(float v, int act) {
  if (act == ACT_SOFTPLUS) return fmaxf(v, 0.f) + log1pf(expf(-fabsf(v)));
  if (act == ACT_GELU) {
    float x3 = v * v * v;   // JAX default gelu: tanh approximation
    return 0.5f * v * (1.f + tanhf(0.7978845608028654f * (v + 0.044715f * x3)));
  }
  if (act == ACT_SIGMOID) return 1.f / (1.f + expf(-v));
  return v;
}
__device__ __forceinline__ float silu_f(float x) { return x / (1.f + expf(-x)); }

// ---------------------------------------------------------------------------
// Generic WMMA f32 GEMM:  C = act(A(MxK) @ B(KxN) + bias) [+ R]
// One wave per 16x16 output tile; 4 waves/block stacked along M.
// Fragment layouts per CDNA5 ISA 7.12.2:
//   A 16x4 : lanes 0-15 -> M=lane, VGPR0/1 = K0/K1; lanes 16-31 -> K2/K3
//   B 4x16 : mirrored per-lane by column
//   C/D    : 8 VGPRs, M = r + 8*(lane>=16), N = lane&15
// ---------------------------------------------------------------------------
__global__ __launch_bounds__(128) void k_gemm(
    const float* __restrict__ A, const float* __restrict__ B,
    const float* __restrict__ bias, const float* __restrict__ R,
    float* __restrict__ C,
    int M, int N, int K, int lda, int ldb, int ldc, int act)
{
  int lane = threadIdx.x & 31;
  int wave = threadIdx.x >> 5;
  int m0 = (blockIdx.y * 4 + wave) * 16;
  int n0 = blockIdx.x * 16;
  if (m0 >= M) return;                       // wave-uniform (never hits: M%64==0)

  int l  = lane & 15;
  int hb = lane >> 4;                        // half-wave: 0 -> K0/K1, 1 -> K2/K3
  int arow  = m0 + l;                        // always < M (M%16==0, exact grid)
  int bcol  = n0 + l;
  int bcolc = (bcol < N) ? bcol : (N - 1);   // clamp: loads stay in bounds

  // A element pair is contiguous & 8B aligned (lda even everywhere).
  const float* Ap = A + (size_t)arow * lda + 2 * hb;
  const float* Bp = B + (size_t)(2 * hb) * ldb + bcolc;

  v8f acc = {0.f, 0.f, 0.f, 0.f, 0.f, 0.f, 0.f, 0.f};

  for (int k0 = 0; k0 < K; k0 += 4) {
    v2f a = *(const v2f*)Ap;                 // global_load_b64
    v2f b;
    b[0] = Bp[0];                            // global_load_b32
    b[1] = Bp[ldb];                          // global_load_b32
    acc = __builtin_amdgcn_wmma_f32_16x16x4_f32(
        /*neg_a=*/false, a, /*neg_b=*/false, b,
        /*c_mod=*/(short)0, acc, /*reuse_a=*/false, /*reuse_b=*/false);
    Ap += 4;
    Bp += (size_t)4 * ldb;
  }

  if (bcol < N) {
    float bv = bias ? bias[bcol] : 0.f;
#pragma unroll
    for (int r = 0; r < 8; ++r) {
      int row = m0 + r + 8 * hb;
      float v = apply_act(acc[r] + bv, act);
      if (R) v += R[(size_t)row * ldc + bcol];
      C[(size_t)row * ldc + bcol] = v;
    }
  }
}

// ----------------------------- small kernels -------------------------------

// Per-series mean/std over time (block per (b,c))
__global__ void k_stats(const float* __restrict__ x, float* __restrict__ mean,
                        float* __restrict__ stdv)
{
  __shared__ float s1[256], s2[256];
  int tid = threadIdx.x;
  int b = blockIdx.x >> 5, c = blockIdx.x & 31;
  const float* base = x + (size_t)b * SEQ_LEN * ENC_IN + c;
  float a0 = base[tid * ENC_IN];
  float a1 = base[(tid + 256) * ENC_IN];
  s1[tid] = a0 + a1;
  s2[tid] = a0 * a0 + a1 * a1;
  __syncthreads();
  for (int s = 128; s > 0; s >>= 1) {
    if (tid < s) { s1[tid] += s1[tid + s]; s2[tid] += s2[tid + s]; }
    __syncthreads();
  }
  if (tid == 0) {
    float mu  = s1[0] * (1.f / SEQ_LEN);
    float var = s2[0] * (1.f / SEQ_LEN) - mu * mu;
    mean[blockIdx.x] = mu;
    stdv[blockIdx.x] = sqrtf(var + 1e-5f);
  }
}

// h = irfft(filt, 512)   (1 block x 512 threads)
__global__ void k_irfft(const float* __restrict__ re, const float* __restrict__ im,
                        float* __restrict__ h)
{
  int t = threadIdx.x;
  const float w = 6.283185307179586f / 512.f;
  float acc = re[0];
  for (int k = 1; k < 256; ++k) {
    int tk = (t * k) & 511;            // exact periodic reduction
    float ang = tk * w;
    acc += 2.f * (re[k] * cosf(ang) - im[k] * sinf(ang));
  }
  acc += (t & 1) ? -re[256] : re[256];
  h[t] = acc * (1.f / 512.f);
}

// Circulant matrix M[t][s] = h[(t-s) mod 512]
__global__ void k_build_M(const float* __restrict__ h, float* __restrict__ Mm)
{
  int idx = blockIdx.x * 256 + threadIdx.x;
  int t = idx >> 9, s = idx & 511;
  Mm[idx] = h[(t - s) & 511];
}

// Xn[s][bc] = (x[b,s,c]-mean)/std
__global__ void k_pack_xn(const float* __restrict__ x, const float* __restrict__ mean,
                          const float* __restrict__ stdv, float* __restrict__ Xn)
{
  int idx = blockIdx.x * 256 + threadIdx.x;   // s*256+bc
  int bc = idx & 255, s = idx >> 8;
  int b = bc >> 5, c = bc & 31;
  float v = x[(size_t)b * SEQ_LEN * ENC_IN + s * ENC_IN + c];
  Xn[idx] = (v - mean[bc]) / stdv[bc];
}

// XP[(bc*41+p)*24+j] = F[(p*12+j)*256+bc]
__global__ void k_patchify(const float* __restrict__ F, float* __restrict__ XP)
{
  int idx = blockIdx.x * 256 + threadIdx.x;
  if (idx >= ROWS * PATCH) return;
  int j = idx % PATCH;
  int r = idx / PATCH;
  int p = r % NUM_PATCH, bc = r / NUM_PATCH;
  XP[idx] = F[(p * STRIDE_ + j) * NSERIES + bc];
}

__global__ void k_add_pos(float* __restrict__ H, const float* __restrict__ pos)
{
  int idx = blockIdx.x * 256 + threadIdx.x;   // ROWS*128
  int d = idx & 127, row = idx >> 7;
  int p = row % NUM_PATCH;
  H[idx] += pos[p * D_MODEL + d];
}

// LayerNorm over last dim (128), one row per block
__global__ void k_ln(const float* __restrict__ X, const float* __restrict__ g,
                     const float* __restrict__ b, float* __restrict__ Y)
{
  __shared__ float sb[D_MODEL];
  int row = blockIdx.x, tid = threadIdx.x;
  float v = X[row * D_MODEL + tid];
  sb[tid] = v; __syncthreads();
  for (int s = D_MODEL / 2; s > 0; s >>= 1) {
    if (tid < s) sb[tid] += sb[tid + s];
    __syncthreads();
  }
  float mu = sb[0] * (1.f / D_MODEL);
  __syncthreads();
  float d0 = v - mu;
  sb[tid] = d0 * d0; __syncthreads();
  for (int s = D_MODEL / 2; s > 0; s >>= 1) {
    if (tid < s) sb[tid] += sb[tid + s];
    __syncthreads();
  }
  float r = rsqrtf(sb[0] * (1.f / D_MODEL) + 1e-5f);
  Y[row * D_MODEL + tid] = d0 * r * g[tid] + b[tid];
}

// Depthwise causal conv (width w) + bias + SiLU on the u-half of UZ
__global__ void k_conv_silu(const float* __restrict__ UZ, const float* __restrict__ cw,
                            const float* __restrict__ cb, float* __restrict__ U, int w)
{
  int row = blockIdx.x;              // 0..ROWS-1
  int d = threadIdx.x;               // 0..255
  int p = row % NUM_PATCH, bc = row / NUM_PATCH;
  float acc = cb[d];
  for (int j = 0; j < w; ++j) {
    int pp = p - (w - 1) + j;
    if (pp >= 0) acc += UZ[(size_t)(bc * NUM_PATCH + pp) * (2 * D_INNER) + d] * cw[j * D_INNER + d];
  }
  U[(size_t)row * D_INNER + d] = silu_f(acc);
}

// Selective scan: thread = (bc, d); loops over L=41, 16 states
__global__ void k_scan(const float* __restrict__ DELTA, const float* __restrict__ U,
                       const float* __restrict__ DBC, const float* __restrict__ A_log,
                       const float* __restrict__ Dp, float* __restrict__ Y)
{
  int bc = blockIdx.x;   // 0..255
  int d = threadIdx.x;   // 0..255
  float Ac[D_STATE], h[D_STATE];
#pragma unroll
  for (int n = 0; n < D_STATE; ++n) { Ac[n] = -expf(A_log[d * D_STATE + n]); h[n] = 0.f; }
  float Dd = Dp[d];
  for (int p = 0; p < NUM_PATCH; ++p) {
    int row = bc * NUM_PATCH + p;
    float delta = DELTA[(size_t)row * D_INNER + d];
    float u = U[(size_t)row * D_INNER + d];
    float du = delta * u;
    const float* dbc = DBC + (size_t)row * DBC_COLS;
    float y = 0.f;
#pragma unroll
    for (int n = 0; n < D_STATE; ++n) {
      float dA = expf(delta * Ac[n]);
      h[n] = dA * h[n] + du * dbc[DT_RANK + n];
      y += h[n] * dbc[DT_RANK + D_STATE + n];
    }
    Y[(size_t)row * D_INNER + d] = y + u * Dd;
  }
}

// Y *= silu(z)  (z = second half of UZ)
__global__ void k_silu_gate(float* __restrict__ Y, const float* __restrict__ UZ)
{
  int row = blockIdx.x;
  int d = threadIdx.x;
  float z = UZ[(size_t)row * (2 * D_INNER) + D_INNER + d];
  Y[(size_t)row * D_INNER + d] *= silu_f(z);
}

// flip along patch axis (rows of 128)
__global__ void k_flip(const float* __restrict__ X, float* __restrict__ Y)
{
  int idx = blockIdx.x * 256 + threadIdx.x;  // ROWS*128
  int d = idx & 127, row = idx >> 7;
  int p = row % NUM_PATCH, bc = row / NUM_PATCH;
  Y[idx] = X[(size_t)(bc * NUM_PATCH + (NUM_PATCH - 1 - p)) * D_MODEL + d];
}

// GATEIN = concat(FF, FR) along last dim
__global__ void k_concat(const float* __restrict__ FF, const float* __restrict__ FR,
                         float* __restrict__ G)
{
  int idx = blockIdx.x * 256 + threadIdx.x;  // ROWS*256
  int col = idx & 255, row = idx >> 8;
  G[idx] = (col < D_MODEL) ? FF[(size_t)row * D_MODEL + col]
                           : FR[(size_t)row * D_MODEL + (col - D_MODEL)];
}

// FUSED = g*FF + (1-g)*FR
__global__ void k_fuse(const float* __restrict__ G, const float* __restrict__ FF,
                       const float* __restrict__ FR, float* __restrict__ O)
{
  int idx = blockIdx.x * 256 + threadIdx.x;  // ROWS*128
  float g = G[idx];
  O[idx] = g * FF[idx] + (1.f - g) * FR[idx];
}

// out[b,t,c] = DEC[bc,t]*std[bc] + mean[bc]
__global__ void k_denorm(const float* __restrict__ DEC, const float* __restrict__ mean,
                         const float* __restrict__ stdv, float* __restrict__ out)
{
  int idx = blockIdx.x * 256 + threadIdx.x;  // 8*96*32
  int c = idx & 31;
  int t = (idx >> 5) % PRED_LEN;
  int b = idx / (PRED_LEN * ENC_IN);
  int bc = b * ENC_IN + c;
  out[idx] = DEC[bc * PRED_LEN + t] * stdv[bc] + mean[bc];
}

// ----------------------------- host side -----------------------------------

// Param leaf indices, assuming jax.tree flatten order (sorted dict keys):
// 0..3: x_enc, x_mark_enc, x_dec, x_mark_dec
// 4 affb_imag, 5 affb_real, 6 enc_norm_b, 7 enc_norm_g
// per-layer (32 leaves, base = 8 + 32*L):
//   +0 gate_b, +1 gate_w,
//   +2.. mamba_global {A_log,D,conv_b,conv_w,dt_b,dt_w,in_proj,out_proj,x_proj}
//   +11.. mamba_local (same 9)
//   +20 norm_b, +21 norm_g, +22 proj_b, +23 proj_w,
//   +24 fc1_b, +25 fc1_w, +26 fc2_b, +27 fc2_w, +28 ffn_norm_b, +29 ffn_norm_g,
//   +30 norm1_b, +31 norm1_g
// 72 pe_b, 73 pe_w, 74 pos_emb, 75 proj_b, 76 proj_w
enum {
  I_XENC = 0, I_AFFB_IM = 4, I_AFFB_RE = 5, I_ENORM_B = 6, I_ENORM_G = 7,
  I_L0 = 8, L_STRIDE = 32,
  L_GATE_B = 0, L_GATE_W = 1, L_MG = 2, L_ML = 11,
  M_ALOG = 0, M_D = 1, M_CONV_B = 2, M_CONV_W = 3, M_DT_B = 4, M_DT_W = 5,
  M_INPROJ = 6, M_OUTPROJ = 7, M_XPROJ = 8,
  L_NORM_B = 20, L_NORM_G = 21, L_PROJ_B = 22, L_PROJ_W = 23,
  L_FC1_B = 24, L_FC1_W = 25, L_FC2_B = 26, L_FC2_W = 27,
  L_FFNN_B = 28, L_FFNN_G = 29, L_NORM1_B = 30, L_NORM1_G = 31,
  I_PE_B = 72, I_PE_W = 73, I_POS = 74, I_PROJ_B = 75, I_PROJ_W = 76
};

struct MambaRefs {
  const float *A_log, *D, *conv_b, *conv_w, *dt_b, *dt_w, *in_proj, *out_proj, *x_proj;
  int dconv;
};

static inline void gemm(hipStream_t s, const float* A, const float* B,
                        const float* bias, const float* R, float* C,
                        int M, int N, int K, int lda, int ldb, int ldc, int act)
{
  dim3 grid((N + 15) / 16, (M + 63) / 64);
  k_gemm<<<grid, 128, 0, s>>>(A, B, bias, R, C, M, N, K, lda, ldb, ldc, act);
}

static inline void run_mamba(hipStream_t s, const MambaRefs& p, const float* Xin,
                             float* Out, bool accumulate,
                             float* UZ, float* U, float* DBC, float* DELTA, float* Y)
{
  // UZ = Xin @ in_proj  (ROWS x 512)
  gemm(s, Xin, p.in_proj, nullptr, nullptr, UZ,
       ROWS, 2 * D_INNER, D_MODEL, D_MODEL, 2 * D_INNER, 2 * D_INNER, ACT_NONE);
  // depthwise causal conv + SiLU on u-half
  k_conv_silu<<<ROWS, D_INNER, 0, s>>>(UZ, p.conv_w, p.conv_b, U, p.dconv);
  // DBC = U @ x_proj  (ROWS x 40)
  gemm(s, U, p.x_proj, nullptr, nullptr, DBC,
       ROWS, DBC_COLS, D_INNER, D_INNER, DBC_COLS, DBC_COLS, ACT_NONE);
  // DELTA = softplus(DBC[:, :8] @ dt_w + dt_b)
  gemm(s, DBC, p.dt_w, p.dt_b, nullptr, DELTA,
       ROWS, D_INNER, DT_RANK, DBC_COLS, D_INNER, D_INNER, ACT_SOFTPLUS);
  // selective scan
  k_scan<<<NSERIES, D_INNER, 0, s>>>(DELTA, U, DBC, p.A_log, p.D, Y);
  // Y *= silu(z)
  k_silu_gate<<<ROWS, D_INNER, 0, s>>>(Y, UZ);
  // Out (+)= Y @ out_proj
  gemm(s, Y, p.out_proj, nullptr, accumulate ? Out : nullptr, Out,
       ROWS, D_MODEL, D_INNER, D_INNER, D_MODEL, D_MODEL, ACT_NONE);
}

static inline MambaRefs mamba_refs(void* const* d_in, int base, int dconv)
{
  MambaRefs r;
  r.A_log    = (const float*)d_in[base + M_ALOG];
  r.D        = (const float*)d_in[base + M_D];
  r.conv_b   = (const float*)d_in[base + M_CONV_B];
  r.conv_w   = (const float*)d_in[base + M_CONV_W];
  r.dt_b     = (const float*)d_in[base + M_DT_B];
  r.dt_w     = (const float*)d_in[base + M_DT_W];
  r.in_proj  = (const float*)d_in[base + M_INPROJ];
  r.out_proj = (const float*)d_in[base + M_OUTPROJ];
  r.x_proj   = (const float*)d_in[base + M_XPROJ];
  r.dconv    = dconv;
  return r;
}

extern "C" void kernel_launch(void* const* d_in, const int* in_sizes, int n_in,
                              void* d_out, int out_size, void* d_ws, size_t ws_size,
                              hipStream_t stream)
{
  (void)in_sizes; (void)n_in; (void)out_size; (void)ws_size;
  const float* x_enc = (const float*)d_in[I_XENC];
  float* W = (float*)d_ws;
  float* out = (float*)d_out;

  // workspace layout (floats); total ~= 28.1M floats ~= 112 MB
  size_t o = 0;
  auto alloc = [&](size_t n) { float* p = W + o; o += n; return p; };
  float* MEAN  = alloc(NSERIES);
  float* STD   = alloc(NSERIES);
  float* Hf    = alloc(SEQ_LEN);
  float* Mm    = alloc((size_t)SEQ_LEN * SEQ_LEN);
  float* XNp   = alloc((size_t)SEQ_LEN * NSERIES);
  float* Ff    = alloc((size_t)SEQ_LEN * NSERIES);
  float* XP    = alloc((size_t)ROWS * PATCH);
  float* H     = alloc((size_t)ROWS * D_MODEL);
  float* HN    = alloc((size_t)ROWS * D_MODEL);
  float* XN    = alloc((size_t)ROWS * D_MODEL);
  float* XR    = alloc((size_t)ROWS * D_MODEL);
  float* FF    = alloc((size_t)ROWS * D_MODEL);
  float* FR    = alloc((size_t)ROWS * D_MODEL);
  float* GATE  = alloc((size_t)ROWS * D_MODEL);   // also reverse-dir accumulator
  float* FUSED = alloc((size_t)ROWS * D_MODEL);
  float* G2    = alloc((size_t)ROWS * 2 * D_MODEL); // gate-in / ffn hidden
  float* UZ    = alloc((size_t)ROWS * 2 * D_INNER);
  float* U     = alloc((size_t)ROWS * D_INNER);
  float* DBCb  = alloc((size_t)ROWS * DBC_COLS);
  float* DELTA = alloc((size_t)ROWS * D_INNER);
  float* Y     = alloc((size_t)ROWS * D_INNER);
  float* DEC   = alloc((size_t)NSERIES * PRED_LEN);

  // 1) per-series stats + normalize + spectral filter (as circulant GEMM)
  k_stats<<<NSERIES, 256, 0, stream>>>(x_enc, MEAN, STD);
  k_irfft<<<1, SEQ_LEN, 0, stream>>>((const float*)d_in[I_AFFB_RE],
                                     (const float*)d_in[I_AFFB_IM], Hf);
  k_build_M<<<(SEQ_LEN * SEQ_LEN) / 256, 256, 0, stream>>>(Hf, Mm);
  k_pack_xn<<<(SEQ_LEN * NSERIES) / 256, 256, 0, stream>>>(x_enc, MEAN, STD, XNp);
  gemm(stream, Mm, XNp, nullptr, nullptr, Ff,
       SEQ_LEN, NSERIES, SEQ_LEN, SEQ_LEN, NSERIES, NSERIES, ACT_NONE);

  // 2) patchify + patch embedding + positional embedding
  k_patchify<<<(ROWS * PATCH + 255) / 256, 256, 0, stream>>>(Ff, XP);
  gemm(stream, XP, (const float*)d_in[I_PE_W], (const float*)d_in[I_PE_B],
       nullptr, H, ROWS, D_MODEL, PATCH, PATCH, D_MODEL, D_MODEL, ACT_NONE);
  k_add_pos<<<(ROWS * D_MODEL) / 256, 256, 0, stream>>>(H, (const float*)d_in[I_POS]);

  // 3) encoder layers
  for (int L = 0; L < 2; ++L) {
    int base = I_L0 + L_STRIDE * L;
    MambaRefs ml = mamba_refs(d_in, base + L_ML, 4);
    MambaRefs mg = mamba_refs(d_in, base + L_MG, 8);

    // HN = LN(H, norm1); XN = LN(HN, bi.norm)
    k_ln<<<ROWS, D_MODEL, 0, stream>>>(H, (const float*)d_in[base + L_NORM1_G],
                                       (const float*)d_in[base + L_NORM1_B], HN);
    k_ln<<<ROWS, D_MODEL, 0, stream>>>(HN, (const float*)d_in[base + L_NORM_G],
                                       (const float*)d_in[base + L_NORM_B], XN);

    // forward direction: FF = mamba_local(XN) + mamba_global(XN)
    run_mamba(stream, ml, XN, FF, false, UZ, U, DBCb, DELTA, Y);
    run_mamba(stream, mg, XN, FF, true,  UZ, U, DBCb, DELTA, Y);

    // reverse direction: FR = flip(mamba_local(flip) + mamba_global(flip))
    k_flip<<<(ROWS * D_MODEL) / 256, 256, 0, stream>>>(XN, XR);
    run_mamba(stream, ml, XR, GATE /*tmp*/, false, UZ, U, DBCb, DELTA, Y);
    run_mamba(stream, mg, XR, GATE /*tmp*/, true,  UZ, U, DBCb, DELTA, Y);
    k_flip<<<(ROWS * D_MODEL) / 256, 256, 0, stream>>>(GATE, FR);

    // gate = sigmoid(concat(FF,FR) @ gate_w + gate_b); fuse; project; residual
    k_concat<<<(ROWS * 2 * D_MODEL) / 256, 256, 0, stream>>>(FF, FR, G2);
    gemm(stream, G2, (const float*)d_in[base + L_GATE_W],
         (const float*)d_in[base + L_GATE_B], nullptr, GATE,
         ROWS, D_MODEL, 2 * D_MODEL, 2 * D_MODEL, D_MODEL, D_MODEL, ACT_SIGMOID);
    k_fuse<<<(ROWS * D_MODEL) / 256, 256, 0, stream>>>(GATE, FF, FR, FUSED);
    gemm(stream, FUSED, (const float*)d_in[base + L_PROJ_W],
         (const float*)d_in[base + L_PROJ_B], H /*residual*/, H,
         ROWS, D_MODEL, D_MODEL, D_MODEL, D_MODEL, D_MODEL, ACT_NONE);

    // FFN with residual
    k_ln<<<ROWS, D_MODEL, 0, stream>>>(H, (const float*)d_in[base + L_FFNN_G],
                                       (const float*)d_in[base + L_FFNN_B], HN);
    gemm(stream, HN, (const float*)d_in[base + L_FC1_W],
         (const float*)d_in[base + L_FC1_B], nullptr, G2,
         ROWS, 2 * D_MODEL, D_MODEL, D_MODEL, 2 * D_MODEL, 2 * D_MODEL, ACT_GELU);
    gemm(stream, G2, (const float*)d_in[base + L_FC2_W],
         (const float*)d_in[base + L_FC2_B], H /*residual*/, H,
         ROWS, D_MODEL, 2 * D_MODEL, 2 * D_MODEL, D_MODEL, D_MODEL, ACT_NONE);
  }

  // 4) final norm + head + de-normalize
  k_ln<<<ROWS, D_MODEL, 0, stream>>>(H, (const float*)d_in[I_ENORM_G],
                                     (const float*)d_in[I_ENORM_B], HN);
  // HN viewed as (256, 5248) row-major
  gemm(stream, HN, (const float*)d_in[I_PROJ_W], (const float*)d_in[I_PROJ_B],
       nullptr, DEC, NSERIES, PRED_LEN, NUM_PATCH * D_MODEL,
       NUM_PATCH * D_MODEL, PRED_LEN, PRED_LEN, ACT_NONE);
  k_denorm<<<(BATCH * PRED_LEN * ENC_IN) / 256, 256, 0, stream>>>(DEC, MEAN, STD, out);
}